// RoiPoolingConvSingle_58171037057307
// MI455X (gfx1250) — compile-verified
//
#include <hip/hip_runtime.h>
#include <math.h>

// ROI pooling (single ROI, SCALE=1): out[c,i,j] = img[0, iy[i], ix[j], c]
//   img: (1, 512, 512, 256) f32 NHWC     roi: (1,5) f32 [_, x0, y0, x1, y1]
//   out: (1, 256, 7, 7) f32
//
// Latency-bound gather (~100 KB total traffic ~= 8 ns at 23.3 TB/s HBM).
// One block per pooled pixel; the block's 256 channel loads are issued as
// CDNA5 async global->LDS copies (ASYNCcnt path), drained with
// s_wait_asynccnt, then transposed out of LDS.

#define POOL      7
#define CHANNELS  256
#define IMG_W     512

__global__ void __launch_bounds__(CHANNELS)
roi_pool_gather_kernel(const float* __restrict__ img,
                       const float* __restrict__ roi,
                       float* __restrict__ out) {
    const int p = blockIdx.x;          // pooled pixel 0..48
    const int i = p / POOL;            // pooled row
    const int j = p - i * POOL;        // pooled col
    const int c = threadIdx.x;         // channel 0..255

    // --- ROI decode (uniform across the whole grid; jnp.round == rintf RNE) ---
    const float r_x0 = roi[1];
    const float r_y0 = roi[2];
    const float r_x1 = roi[3];
    const float r_y1 = roi[4];
    const int x_min = (int)rintf(r_x0);
    const int y_min = (int)rintf(r_y0);
    const int x_max = (int)rintf(r_x1);
    const int y_max = (int)rintf(r_y1);

    // Match reference arithmetic exactly: f32 divide, f32 multiply, floor.
    const float h = (float)(y_max - y_min + 1);
    const float w = (float)(x_max - x_min + 1);
    const float hs = h / (float)POOL;
    const float ws = w / (float)POOL;

    int iy = y_min + (int)floorf((float)i * hs);
    int ix = x_min + (int)floorf((float)j * ws);
    iy = min(max(iy, y_min), y_max);
    ix = min(max(ix, x_min), x_max);

    // --- Async global -> LDS copy of this pixel's 256-float channel row ---
    __shared__ float tile[CHANNELS];

    const float* gp = img + ((size_t)iy * IMG_W + (size_t)ix) * CHANNELS + c;
    const unsigned long long gaddr = (unsigned long long)(uintptr_t)gp;
    // Flat addr of a __shared__ object: low 32 bits == allocation-relative LDS
    // byte offset (the address truncated to 32 bits) — exactly what the async
    // instruction's VDST operand wants (an offset added to the wave's LDS base).
    const unsigned int lds_off = (unsigned int)(uintptr_t)(&tile[c]);

    // GV mode: global_load_async_to_lds_b32 vdst(lds offset), vaddr(64b), off
    asm volatile("global_load_async_to_lds_b32 %0, %1, off"
                 :
                 : "v"(lds_off), "v"(gaddr)
                 : "memory");
    // Drain this wave's ASYNCcnt, then sync the workgroup (8 wave32s).
    asm volatile("s_wait_asynccnt 0" ::: "memory");
    __syncthreads();

    // --- Transposed store: NHWC tile row -> NCHW output ---
    out[(size_t)c * (POOL * POOL) + p] = tile[c];
}

extern "C" void kernel_launch(void* const* d_in, const int* in_sizes, int n_in,
                              void* d_out, int out_size, void* d_ws, size_t ws_size,
                              hipStream_t stream) {
    const float* img = (const float*)d_in[0];   // (1,512,512,256) f32
    const float* roi = (const float*)d_in[1];   // (1,5) f32
    float* out = (float*)d_out;                 // (1,256,7,7) f32 = 12544

    (void)in_sizes; (void)n_in; (void)out_size; (void)d_ws; (void)ws_size;

    roi_pool_gather_kernel<<<dim3(POOL * POOL), dim3(CHANNELS), 0, stream>>>(
        img, roi, out);
}